// EGAT_55130200211689
// MI455X (gfx1250) — compile-verified
//
#include <hip/hip_runtime.h>
#include <math.h>

// ---------------- problem constants (match reference) ----------------
#define NNODES   262144      // 256 * 1024
#define NEDGES   8388608
#define BATCH    256
#define FDIM     4096
#define NEG_SLOPE 0.2f
#define BN_EPS    1e-5f

typedef __attribute__((ext_vector_type(2))) float v2f;
typedef __attribute__((ext_vector_type(8))) float v8f;

// =====================================================================
// Kernel A: node transform.
//   h0[n] = [ emb1(x[n,7:16]), x[n,0:7] ]  (8 features)
//   h_c[n] = h0[n] @ W_c (8->4) for the 3 attention channels.
// One wave = one 16-node tile. A = h0 tile (16x8, split into two K=4
// chunks), B = [W0 | W1 | W2 | 0] (8x16).  Two V_WMMA_F32_16X16X4_F32
// produce all 3 channels' h for 16 nodes at once.
// =====================================================================
__global__ __launch_bounds__(256)
void egat_node_transform(const float* __restrict__ x,
                         const float* __restrict__ w_emb1,
                         const float* __restrict__ b_emb1,
                         const float* __restrict__ w_gat,
                         float* __restrict__ hbuf /* [3][N][4] */) {
    __shared__ float h0s[8][16][8];            // per-wave 16x8 tile
    const int lane = threadIdx.x & 31;
    const int wv   = threadIdx.x >> 5;
    const int m    = lane & 15;                // row / col within tile
    const int g    = lane >> 4;                // half-wave id
    const int nodebase = (blockIdx.x * 8 + wv) * 16;
    const int n = nodebase + m;

    // Build h0 tile cooperatively: half 0 does features 0..3 (feat 0 = emb1),
    // half 1 does features 4..7.
    if (g == 0) {
        float f0 = b_emb1[0];
#pragma unroll
        for (int k = 0; k < 9; ++k) f0 += x[n * 16 + 7 + k] * w_emb1[k];
        h0s[wv][m][0] = f0;
        h0s[wv][m][1] = x[n * 16 + 0];
        h0s[wv][m][2] = x[n * 16 + 1];
        h0s[wv][m][3] = x[n * 16 + 2];
    } else {
        h0s[wv][m][4] = x[n * 16 + 3];
        h0s[wv][m][5] = x[n * 16 + 4];
        h0s[wv][m][6] = x[n * 16 + 5];
        h0s[wv][m][7] = x[n * 16 + 6];
    }
    __syncthreads();

    // A fragment (16x4 f32): lane L, vgpr v -> M = L%16, K = 2*(L/16)+v
    v2f a0, a1, b0, b1;
    a0.x = h0s[wv][m][2 * g + 0];
    a0.y = h0s[wv][m][2 * g + 1];
    a1.x = h0s[wv][m][4 + 2 * g + 0];
    a1.y = h0s[wv][m][4 + 2 * g + 1];

    // B fragment (4x16 f32): lane L, vgpr v -> N = L%16, K = 2*(L/16)+v
    // Column layout: col = 4*channel + j ; cols 12..15 are zero pad.
    const int  ncol  = m;
    const bool valid = ncol < 12;
    const int  ch = ncol >> 2, jj = ncol & 3;
    b0.x = valid ? w_gat[ch * 32 + (2 * g + 0) * 4 + jj] : 0.f;
    b0.y = valid ? w_gat[ch * 32 + (2 * g + 1) * 4 + jj] : 0.f;
    b1.x = valid ? w_gat[ch * 32 + (4 + 2 * g + 0) * 4 + jj] : 0.f;
    b1.y = valid ? w_gat[ch * 32 + (4 + 2 * g + 1) * 4 + jj] : 0.f;

    v8f c = {};
    c = __builtin_amdgcn_wmma_f32_16x16x4_f32(false, a0, false, b0, (short)0, c, false, false);
    c = __builtin_amdgcn_wmma_f32_16x16x4_f32(false, a1, false, b1, (short)0, c, false, false);

    // D fragment: lane L, vgpr v -> row M = v + 8*(L/16), col = L%16
    if (valid) {
#pragma unroll
        for (int v = 0; v < 8; ++v) {
            int M = v + 8 * g;
            hbuf[((size_t)ch * NNODES + nodebase + M) * 4 + jj] = c[v];
        }
    }
}

// ---------------------------------------------------------------------
// order-preserving f32 <-> u32 key for atomicMax-based segment max
// ---------------------------------------------------------------------
__device__ __forceinline__ unsigned f2key(float x) {
    unsigned k = __float_as_uint(x);
    return (k & 0x80000000u) ? ~k : (k | 0x80000000u);
}
__device__ __forceinline__ float key2f(unsigned k) {
    return (k & 0x80000000u) ? __uint_as_float(k & 0x7FFFFFFFu)
                             : __uint_as_float(~k);
}

// =====================================================================
// Kernel B1: edge pass 1 — leaky-relu logits, segment max over dst.
// h table (12 MB) is L2-resident, so the gathers hit L2.
// =====================================================================
__global__ __launch_bounds__(256)
void egat_edge_max(const int* __restrict__ ei,
                   const float* __restrict__ eattr,
                   const float* __restrict__ att,
                   const float* __restrict__ hbuf,
                   unsigned* __restrict__ mkey) {
    const float4* h4 = (const float4*)hbuf;
    const int stride = gridDim.x * blockDim.x;
    for (int e = blockIdx.x * blockDim.x + threadIdx.x; e < NEDGES; e += stride) {
        const int s = ei[e];
        const int d = ei[NEDGES + e];
#pragma unroll
        for (int c = 0; c < 3; ++c) {
            float4 hs = h4[c * NNODES + s];
            float4 hd = h4[c * NNODES + d];
            const float* a = att + c * 9;
            float l = hd.x * a[0] + hd.y * a[1] + hd.z * a[2] + hd.w * a[3]
                    + hs.x * a[4] + hs.y * a[5] + hs.z * a[6] + hs.w * a[7]
                    + eattr[e * 3 + c] * a[8];
            l = (l > 0.f) ? l : NEG_SLOPE * l;
            atomicMax(&mkey[c * NNODES + d], f2key(l));
        }
    }
}

// =====================================================================
// Kernel B2: edge pass 2 — ex = exp(logit - max), fused accumulation of
// numerator (ex * hs) and denominator (ex).  alpha never materialized:
// out = num / den at the node level afterwards.
// =====================================================================
__global__ __launch_bounds__(256)
void egat_edge_sum(const int* __restrict__ ei,
                   const float* __restrict__ eattr,
                   const float* __restrict__ att,
                   const float* __restrict__ hbuf,
                   const unsigned* __restrict__ mkey,
                   float* __restrict__ den,
                   float* __restrict__ num) {
    const float4* h4 = (const float4*)hbuf;
    const int stride = gridDim.x * blockDim.x;
    for (int e = blockIdx.x * blockDim.x + threadIdx.x; e < NEDGES; e += stride) {
        const int s = ei[e];
        const int d = ei[NEDGES + e];
#pragma unroll
        for (int c = 0; c < 3; ++c) {
            float4 hs = h4[c * NNODES + s];
            float4 hd = h4[c * NNODES + d];
            const float* a = att + c * 9;
            float l = hd.x * a[0] + hd.y * a[1] + hd.z * a[2] + hd.w * a[3]
                    + hs.x * a[4] + hs.y * a[5] + hs.z * a[6] + hs.w * a[7]
                    + eattr[e * 3 + c] * a[8];
            l = (l > 0.f) ? l : NEG_SLOPE * l;
            float mx = key2f(mkey[c * NNODES + d]);
            float ex = __expf(l - mx);
            atomicAdd(&den[c * NNODES + d], ex);
            float* nb = &num[((size_t)c * NNODES + d) * 4];
            atomicAdd(nb + 0, ex * hs.x);
            atomicAdd(nb + 1, ex * hs.y);
            atomicAdd(nb + 2, ex * hs.z);
            atomicAdd(nb + 3, ex * hs.w);
        }
    }
}

// =====================================================================
// Kernel C: per-node fuse: fused[n][j] = b_emb2 + sum_c (num/den)*w_emb2[c].
// fused stored row-major (N,4) == flat (B, 4096) directly.
// =====================================================================
__global__ __launch_bounds__(256)
void egat_fuse(const float* __restrict__ num,
               const float* __restrict__ den,
               const float* __restrict__ w_emb2,
               const float* __restrict__ b_emb2,
               float* __restrict__ flat) {
    const int n = blockIdx.x * blockDim.x + threadIdx.x;
    if (n >= NNODES) return;
    const float bb = b_emb2[0];
    float4 acc = make_float4(bb, bb, bb, bb);
    const float4* num4 = (const float4*)num;
#pragma unroll
    for (int c = 0; c < 3; ++c) {
        float  dv = fmaxf(den[c * NNODES + n], 1e-16f);
        float  wc = w_emb2[c] / dv;
        float4 nm = num4[c * NNODES + n];
        acc.x += nm.x * wc;
        acc.y += nm.y * wc;
        acc.z += nm.z * wc;
        acc.w += nm.w * wc;
    }
    ((float4*)flat)[n] = acc;
}

// =====================================================================
// Kernel D: fc1 GEMM  (256 x 4096) @ (4096 x 4) with f32 WMMA, split-K.
// One wave per block; blockIdx.x = 16-row tile, blockIdx.y = K split.
// Partial 16x16 tiles (cols 0..3 valid) reduced with global f32 atomics.
// =====================================================================
__global__ __launch_bounds__(32)
void fc1_wmma(const float* __restrict__ flat,
              const float* __restrict__ w_fc1,
              float* __restrict__ z_pre /* [256][4] */) {
    const int lane = threadIdx.x;
    const int m = lane & 15, g = lane >> 4;
    const int row  = blockIdx.x * 16 + m;
    const int ncol = m;
    const int kbeg = blockIdx.y * (FDIM / 8);
    const int kend = kbeg + (FDIM / 8);

    v8f c = {};
    for (int k0 = kbeg; k0 < kend; k0 += 4) {
        v2f a, b;
        a.x = flat[row * FDIM + k0 + 2 * g + 0];
        a.y = flat[row * FDIM + k0 + 2 * g + 1];
        b.x = (ncol < 4) ? w_fc1[(k0 + 2 * g + 0) * 4 + ncol] : 0.f;
        b.y = (ncol < 4) ? w_fc1[(k0 + 2 * g + 1) * 4 + ncol] : 0.f;
        c = __builtin_amdgcn_wmma_f32_16x16x4_f32(false, a, false, b, (short)0, c, false, false);
    }
    if (ncol < 4) {
#pragma unroll
        for (int v = 0; v < 8; ++v) {
            int M = v + 8 * g;
            atomicAdd(&z_pre[(blockIdx.x * 16 + M) * 4 + ncol], c[v]);
        }
    }
}

// =====================================================================
// Kernel E: head — bias + relu + batch-norm over the 256 rows + fc4.
// Single 256-thread block; one thread per batch row.
// =====================================================================
__global__ __launch_bounds__(256)
void egat_head(const float* __restrict__ z_pre,
               const float* __restrict__ b_fc1,
               const float* __restrict__ bn_gamma,
               const float* __restrict__ bn_beta,
               const float* __restrict__ w_fc4,
               const float* __restrict__ b_fc4,
               float* __restrict__ out /* 513 floats: (256,2) + reg */) {
    __shared__ float s_sum[4], s_sq[4];
    const int b = threadIdx.x;
    if (b < 4) { s_sum[b] = 0.f; s_sq[b] = 0.f; }
    __syncthreads();

    float z[4];
#pragma unroll
    for (int j = 0; j < 4; ++j) {
        z[j] = fmaxf(z_pre[b * 4 + j] + b_fc1[j], 0.f);
        atomicAdd(&s_sum[j], z[j]);
        atomicAdd(&s_sq[j], z[j] * z[j]);
    }
    __syncthreads();

    float zn[4];
#pragma unroll
    for (int j = 0; j < 4; ++j) {
        float mu  = s_sum[j] * (1.f / BATCH);
        float var = s_sq[j] * (1.f / BATCH) - mu * mu;
        zn[j] = (z[j] - mu) * rsqrtf(var + BN_EPS) * bn_gamma[j] + bn_beta[j];
    }
#pragma unroll
    for (int q = 0; q < 2; ++q) {
        out[b * 2 + q] = zn[0] * w_fc4[0 * 2 + q] + zn[1] * w_fc4[1 * 2 + q]
                       + zn[2] * w_fc4[2 * 2 + q] + zn[3] * w_fc4[3 * 2 + q]
                       + b_fc4[q];
    }
    if (b == 0) out[BATCH * 2] = 0.f;   // reg output
}

// =====================================================================
// Launch
// =====================================================================
extern "C" void kernel_launch(void* const* d_in, const int* in_sizes, int n_in,
                              void* d_out, int out_size, void* d_ws, size_t ws_size,
                              hipStream_t stream) {
    const float* x        = (const float*)d_in[0];
    const int*   ei       = (const int*)  d_in[1];
    const float* eattr    = (const float*)d_in[2];
    /* y = d_in[3] unused */
    const float* w_emb1   = (const float*)d_in[4];
    const float* b_emb1   = (const float*)d_in[5];
    const float* w_gat    = (const float*)d_in[6];
    const float* att_gat  = (const float*)d_in[7];
    const float* w_emb2   = (const float*)d_in[8];
    const float* b_emb2   = (const float*)d_in[9];
    const float* bn_gamma = (const float*)d_in[10];
    const float* bn_beta  = (const float*)d_in[11];
    const float* w_fc1    = (const float*)d_in[12];
    const float* b_fc1    = (const float*)d_in[13];
    const float* w_fc4    = (const float*)d_in[14];
    const float* b_fc4    = (const float*)d_in[15];
    float* out = (float*)d_out;

    // workspace layout (16B aligned regions)
    char*  ws       = (char*)d_ws;
    size_t OFF_H    = 0;                                      // 3*N*4 f32
    size_t OFF_MKEY = OFF_H    + (size_t)3 * NNODES * 4 * 4;  // 3*N u32
    size_t OFF_DEN  = OFF_MKEY + (size_t)3 * NNODES * 4;      // 3*N f32
    size_t OFF_NUM  = OFF_DEN  + (size_t)3 * NNODES * 4;      // 3*N*4 f32
    size_t OFF_FLAT = OFF_NUM  + (size_t)3 * NNODES * 4 * 4;  // N*4 f32
    size_t OFF_Z    = OFF_FLAT + (size_t)NNODES * 4 * 4;      // 256*4 f32
    size_t END      = OFF_Z    + (size_t)BATCH * 4 * 4;

    float*    hbuf = (float*)   (ws + OFF_H);
    unsigned* mkey = (unsigned*)(ws + OFF_MKEY);
    float*    den  = (float*)   (ws + OFF_DEN);
    float*    num  = (float*)   (ws + OFF_NUM);
    float*    flat = (float*)   (ws + OFF_FLAT);
    float*    zpre = (float*)   (ws + OFF_Z);

    // zero all accumulators (mkey=0 is the encoded minimum key)
    hipMemsetAsync(ws + OFF_MKEY, 0, END - OFF_MKEY, stream);

    // A: node transform (WMMA f32 16x16x4), 16 nodes/wave, 8 waves/block
    egat_node_transform<<<NNODES / (16 * 8), 256, 0, stream>>>(
        x, w_emb1, b_emb1, w_gat, hbuf);

    // B1/B2: two edge passes (segment max, then fused exp/num/den)
    egat_edge_max<<<8192, 256, 0, stream>>>(ei, eattr, att_gat, hbuf, mkey);
    egat_edge_sum<<<8192, 256, 0, stream>>>(ei, eattr, att_gat, hbuf, mkey, den, num);

    // C: softmax divide + channel mix -> flat (B, 4096)
    egat_fuse<<<NNODES / 256, 256, 0, stream>>>(num, den, w_emb2, b_emb2, flat);

    // D: fc1 GEMM with WMMA, 16 row tiles x 8 K-splits
    fc1_wmma<<<dim3(16, 8), 32, 0, stream>>>(flat, w_fc1, zpre);

    // E: bias/relu/batchnorm/fc4 head
    egat_head<<<1, 256, 0, stream>>>(zpre, b_fc1, bn_gamma, bn_beta,
                                     w_fc4, b_fc4, out);
    (void)in_sizes; (void)n_in; (void)out_size; (void)ws_size; (void)END;
}